// transformer_16587163697732
// MI455X (gfx1250) — compile-verified
//
#include <hip/hip_runtime.h>

typedef __attribute__((ext_vector_type(2))) float v2f;
typedef __attribute__((ext_vector_type(4))) float v4f;
typedef __attribute__((ext_vector_type(8))) float v8f;

// ---------------------------------------------------------------------------
// Generic batched GEMM: C[z] = alpha * A[z] (MxK, row major) @ B[z] (KxN, row
// major).  Strides are per-batch element strides (0 => shared operand).
//
// Block: 256 threads = 8 wave32 waves laid out 4x2; block C tile 128x64; each
// wave owns a 32x32 C tile (4 f32 accumulators).  K is staged 32 at a time
// through double-buffered LDS.  A is stored [m][k], B is stored k-major
// [n][k] so every WMMA fragment is one aligned ds_load_b64.  Global loads are
// float4 (b128), prefetched into registers one stage ahead.
// Requires M % 128 == 0, N % 64 == 0, K % 32 == 0 (true for all uses here).
// ---------------------------------------------------------------------------
__global__ __launch_bounds__(256) void gemm_wmma_f32(
    const float* __restrict__ A, const float* __restrict__ B,
    float* __restrict__ C, int M, int N, int K,
    long sA, long sB, long sC, float alpha)
{
    constexpr int BM = 128, BN = 64, BK = 32, PAD = 4;  // stride 36 -> 144B rows
    __shared__ float As[2][BM][BK + PAD];
    __shared__ float Bs[2][BN][BK + PAD];

    const float* Ab = A + (size_t)blockIdx.z * (size_t)sA;
    const float* Bb = B + (size_t)blockIdx.z * (size_t)sB;
    float*       Cb = C + (size_t)blockIdx.z * (size_t)sC;

    const int blockM = blockIdx.y * BM;
    const int blockN = blockIdx.x * BN;
    const int lane = threadIdx.x & 31;
    const int wave = threadIdx.x >> 5;
    const int row0 = (wave & 3) * 32;    // wave's 32-row strip in block tile
    const int col0 = (wave >> 2) * 32;   // wave's 32-col strip in block tile
    const int ml = lane & 15;
    const int kh = (lane >> 4) * 2;      // lanes 16-31 hold K+2/K+3 (ISA 7.12.2)

    // A tile: 128x32 = 1024 float4, 4 per thread.  B tile: 32x64 = 512 float4,
    // 2 per thread (transposed into k-major LDS on store).
    int am[4], ak[4], bk[2], bn[2];
#pragma unroll
    for (int j = 0; j < 4; ++j) {
        const int idx = (int)threadIdx.x + j * 256;
        am[j] = idx >> 3;
        ak[j] = (idx & 7) << 2;
    }
#pragma unroll
    for (int j = 0; j < 2; ++j) {
        const int idx = (int)threadIdx.x + j * 256;
        bk[j] = idx >> 4;
        bn[j] = (idx & 15) << 2;
    }

    v4f ra[4], rb[2];

    // ---- preload stage 0 into LDS buffer 0 ----
#pragma unroll
    for (int j = 0; j < 4; ++j)
        ra[j] = *(const v4f*)&Ab[(size_t)(blockM + am[j]) * K + ak[j]];
#pragma unroll
    for (int j = 0; j < 2; ++j)
        rb[j] = *(const v4f*)&Bb[(size_t)bk[j] * N + blockN + bn[j]];
#pragma unroll
    for (int j = 0; j < 4; ++j)
        *(v4f*)&As[0][am[j]][ak[j]] = ra[j];
#pragma unroll
    for (int j = 0; j < 2; ++j) {
        Bs[0][bn[j] + 0][bk[j]] = rb[j].x;
        Bs[0][bn[j] + 1][bk[j]] = rb[j].y;
        Bs[0][bn[j] + 2][bk[j]] = rb[j].z;
        Bs[0][bn[j] + 3][bk[j]] = rb[j].w;
    }
    __syncthreads();

    v8f acc00 = {}, acc01 = {}, acc10 = {}, acc11 = {};

    const int nStages = K / BK;
    for (int s = 0; s < nStages; ++s) {
        const int buf = s & 1;
        const int nxt = buf ^ 1;

        // Prefetch next K stage into registers (overlaps with WMMA below).
        if (s + 1 < nStages) {
            const int k0 = (s + 1) * BK;
#pragma unroll
            for (int j = 0; j < 4; ++j)
                ra[j] = *(const v4f*)&Ab[(size_t)(blockM + am[j]) * K + k0 + ak[j]];
#pragma unroll
            for (int j = 0; j < 2; ++j)
                rb[j] = *(const v4f*)&Bb[(size_t)(k0 + bk[j]) * N + blockN + bn[j]];
        }

        // 8 K-steps x 4 WMMAs on the current LDS buffer.
#pragma unroll
        for (int kk = 0; kk < BK; kk += 4) {
            const v2f a0 = *(const v2f*)&As[buf][row0 + ml][kk + kh];
            const v2f a1 = *(const v2f*)&As[buf][row0 + 16 + ml][kk + kh];
            const v2f b0 = *(const v2f*)&Bs[buf][col0 + ml][kk + kh];
            const v2f b1 = *(const v2f*)&Bs[buf][col0 + 16 + ml][kk + kh];
            acc00 = __builtin_amdgcn_wmma_f32_16x16x4_f32(
                false, a0, false, b0, (short)0, acc00, false, false);
            acc01 = __builtin_amdgcn_wmma_f32_16x16x4_f32(
                false, a0, false, b1, (short)0, acc01, false, false);
            acc10 = __builtin_amdgcn_wmma_f32_16x16x4_f32(
                false, a1, false, b0, (short)0, acc10, false, false);
            acc11 = __builtin_amdgcn_wmma_f32_16x16x4_f32(
                false, a1, false, b1, (short)0, acc11, false, false);
        }

        // Stage next tile into the alternate LDS buffer (disjoint from reads).
        if (s + 1 < nStages) {
#pragma unroll
            for (int j = 0; j < 4; ++j)
                *(v4f*)&As[nxt][am[j]][ak[j]] = ra[j];
#pragma unroll
            for (int j = 0; j < 2; ++j) {
                Bs[nxt][bn[j] + 0][bk[j]] = rb[j].x;
                Bs[nxt][bn[j] + 1][bk[j]] = rb[j].y;
                Bs[nxt][bn[j] + 2][bk[j]] = rb[j].z;
                Bs[nxt][bn[j] + 3][bk[j]] = rb[j].w;
            }
        }
        __syncthreads();
    }

    // C/D layout: VGPR r -> row r (lanes 0-15) / row 8+r (lanes 16-31), col l&15
    const int r0 = blockM + row0 + ((lane >> 4) << 3);
    const int c0 = blockN + col0 + ml;
#pragma unroll
    for (int r = 0; r < 8; ++r) {
        Cb[(size_t)(r0 + r) * N + c0]           = alpha * acc00[r];
        Cb[(size_t)(r0 + r) * N + c0 + 16]      = alpha * acc01[r];
        Cb[(size_t)(r0 + 16 + r) * N + c0]      = alpha * acc10[r];
        Cb[(size_t)(r0 + 16 + r) * N + c0 + 16] = alpha * acc11[r];
    }
}

// ---------------------------------------------------------------------------
// In-place row softmax for 512-wide rows.  One 256-thread block per row.
// ---------------------------------------------------------------------------
__global__ __launch_bounds__(256) void softmax512(float* __restrict__ data)
{
    float* row = data + (size_t)blockIdx.x * 512;
    __shared__ float red[256];
    const int t = threadIdx.x;
    float v0 = row[t];
    float v1 = row[t + 256];

    red[t] = fmaxf(v0, v1);
    __syncthreads();
    for (int s = 128; s > 0; s >>= 1) {
        if (t < s) red[t] = fmaxf(red[t], red[t + s]);
        __syncthreads();
    }
    const float mx = red[0];
    __syncthreads();

    const float e0 = expf(v0 - mx);
    const float e1 = expf(v1 - mx);
    red[t] = e0 + e1;
    __syncthreads();
    for (int s = 128; s > 0; s >>= 1) {
        if (t < s) red[t] += red[t + s];
        __syncthreads();
    }
    const float inv = 1.0f / red[0];
    row[t]       = e0 * inv;
    row[t + 256] = e1 * inv;
}

// W_sum[s,t] = sum_h W[s, h*512 + t]  (valid for the identical-batch layers)
__global__ __launch_bounds__(256) void wsum_k(const float* __restrict__ W,
                                              float* __restrict__ Ws)
{
    const int idx = blockIdx.x * 256 + threadIdx.x;   // 512*512 total
    const int s = idx >> 9, t = idx & 511;
    float acc = 0.0f;
#pragma unroll
    for (int h = 0; h < 8; ++h) acc += W[(size_t)s * 4096 + h * 512 + t];
    Ws[idx] = acc;
}

// Broadcast shared [512,512] result to all 8 batches of d_out.
__global__ __launch_bounds__(256) void bcast_k(const float* __restrict__ y,
                                               float* __restrict__ out)
{
    const size_t idx = (size_t)blockIdx.x * 256 + threadIdx.x;
    out[idx] = y[idx & 262143];   // 262144 = 512*512
}

extern "C" void kernel_launch(void* const* d_in, const int* in_sizes, int n_in,
                              void* d_out, int out_size, void* d_ws, size_t ws_size,
                              hipStream_t stream)
{
    (void)in_sizes; (void)n_in; (void)out_size; (void)ws_size;

    const float* emb  = (const float*)d_in[0];   // [8,512,512]
    const float* keyW = (const float*)d_in[1];   // [512,512]
    const float* valW = (const float*)d_in[2];   // [512,512]
    const float* qryW = (const float*)d_in[3];   // [512,512]
    const float* Wmat = (const float*)d_in[4];   // [512,4096]

    const int    S  = 512, Bt = 8;
    const size_t PB  = (size_t)S * 512;          // 262144 elems per batch slab
    const size_t BIG = (size_t)Bt * PB;          // 2097152

    float* ws  = (float*)d_ws;
    float* bK  = ws;                 // [8,512,512]
    float* bQ  = ws + BIG;           // [8,512,512]  (also reused for att)
    float* bV  = ws + 2 * BIG;       // [8,512,512]
    float* bS  = ws + 3 * BIG;       // [8,512,512]  scores
    float* bY  = ws + 4 * BIG;       // [512,512]    shared layer output
    float* bWs = bY + PB;            // [512,512]    W_sum

    const float scale = 1.0f / 22.0f;            // 1 / int(sqrt(512))
    dim3 blk(256);
    dim3 gB(8, 4, 8);                            // N/64, M/128, batch
    dim3 g1(8, 4, 1);

    wsum_k<<<(unsigned)(PB / 256), blk, 0, stream>>>(Wmat, bWs);

    // ---- Layer 1: batches are distinct ----
    gemm_wmma_f32<<<gB, blk, 0, stream>>>(emb, keyW, bK, 512, 512, 512,
                                          (long)PB, 0, (long)PB, 1.0f);
    gemm_wmma_f32<<<gB, blk, 0, stream>>>(emb, qryW, bQ, 512, 512, 512,
                                          (long)PB, 0, (long)PB, 1.0f);
    gemm_wmma_f32<<<gB, blk, 0, stream>>>(emb, valW, bV, 512, 512, 512,
                                          (long)PB, 0, (long)PB, 1.0f);
    // scores = (Q @ K) * scale  (reshape quirk => plain GEMM with K as stored)
    gemm_wmma_f32<<<gB, blk, 0, stream>>>(bQ, bK, bS, 512, 512, 512,
                                          (long)PB, (long)PB, (long)PB, scale);
    softmax512<<<Bt * S, blk, 0, stream>>>(bS);
    // att -> bQ (Q no longer needed)
    gemm_wmma_f32<<<gB, blk, 0, stream>>>(bS, bV, bQ, 512, 512, 512,
                                          (long)PB, (long)PB, (long)PB, 1.0f);
    // stack/reshape quirk: y = W[512,4096] @ att_concat[4096,512]; identical
    // for every batch, so compute once.
    gemm_wmma_f32<<<g1, blk, 0, stream>>>(Wmat, bQ, bY, 512, 512, 4096,
                                          0, 0, 0, 1.0f);

    // ---- Layers 2..4: all batches identical -> single-batch GEMMs ----
    for (int layer = 1; layer < 4; ++layer) {
        gemm_wmma_f32<<<g1, blk, 0, stream>>>(bY, keyW, bK, 512, 512, 512,
                                              0, 0, 0, 1.0f);
        gemm_wmma_f32<<<g1, blk, 0, stream>>>(bY, qryW, bQ, 512, 512, 512,
                                              0, 0, 0, 1.0f);
        gemm_wmma_f32<<<g1, blk, 0, stream>>>(bY, valW, bV, 512, 512, 512,
                                              0, 0, 0, 1.0f);
        gemm_wmma_f32<<<g1, blk, 0, stream>>>(bQ, bK, bS, 512, 512, 512,
                                              0, 0, 0, scale);
        softmax512<<<S, blk, 0, stream>>>(bS);
        gemm_wmma_f32<<<g1, blk, 0, stream>>>(bS, bV, bQ, 512, 512, 512,
                                              0, 0, 0, 1.0f);
        gemm_wmma_f32<<<g1, blk, 0, stream>>>(bWs, bQ, bY, 512, 512, 512,
                                              0, 0, 0, 1.0f);
    }

    // ---- Emit [8,512,512] output (identical per batch) ----
    bcast_k<<<(unsigned)((Bt * PB) / 256), blk, 0, stream>>>(bY, (float*)d_out);
}